// RCNNCell_66778151518249
// MI455X (gfx1250) — compile-verified
//
#include <hip/hip_runtime.h>
#include <math.h>

#define TILE  64
#define HALO  12
#define ESZ   (TILE + 2*HALO)   // 88 staged extent
#define ESTR  ESZ               // 88 stride: rows are 352B = 16B-aligned
#define EA    (ESZ * ESTR)      // 7744 floats per staged field (16B multiple)
#define RB    3                 // first valid stage row
#define RROWS (TILE + 18)       // 82 stage rows [3,85)
#define QX    (ESZ / 4)         // 22 x-quads per row, cols [0,88)
#define NTHR  512

constexpr float kDT = 0.5f;
// Laplacian coefficients: LAP/(dx*dx)/180
constexpr float LC0 = (-980.0f / (0.01f * 0.01f)) / 180.0f;
constexpr float LC1 = ( 270.0f / (0.01f * 0.01f)) / 180.0f;
constexpr float LC2 = ( -27.0f / (0.01f * 0.01f)) / 180.0f;
constexpr float LC3 = (   2.0f / (0.01f * 0.01f)) / 180.0f;

// ---- CDNA5 async global->LDS (ASYNCcnt, bypasses VGPRs) -------------------
__device__ __forceinline__ void async_ld_b32(uint32_t lds_byte_addr, const float* gaddr) {
  asm volatile("global_load_async_to_lds_b32 %0, %1, off"
               :: "v"(lds_byte_addr), "v"(gaddr) : "memory");
}
__device__ __forceinline__ void wait_async0() {
#if __has_builtin(__builtin_amdgcn_s_wait_asynccnt)
  __builtin_amdgcn_s_wait_asynccnt(0);
#else
  asm volatile("s_wait_asynccnt 0" ::: "memory");
#endif
}

// 16B-aligned LDS vector access (alignment guaranteed by construction)
__device__ __forceinline__ float4 LD4(const float* __restrict__ s, int c) {
  return *(const float4*)(s + c);
}
__device__ __forceinline__ void ST4(float* __restrict__ s, int c, float4 v) {
  *(float4*)(s + c) = v;
}

// Laplacian for 4 consecutive points at 16B-aligned cb.
// 9 ds_load_b128 total: 3 for the x-window (x-4..x+7), 6 for y taps.
__device__ __forceinline__ void lap_quad(const float* __restrict__ s, int cb,
                                         float lap[4], float ctr[4]) {
  float w[12];
  {
    float4 a = LD4(s, cb - 4), b = LD4(s, cb), c = LD4(s, cb + 4);
    w[0]=a.x; w[1]=a.y; w[2]=a.z; w[3]=a.w;
    w[4]=b.x; w[5]=b.y; w[6]=b.z; w[7]=b.w;
    w[8]=c.x; w[9]=c.y; w[10]=c.z; w[11]=c.w;
  }
  float4 m3 = LD4(s, cb - 3*ESTR), m2 = LD4(s, cb - 2*ESTR), m1 = LD4(s, cb - ESTR);
  float4 p1 = LD4(s, cb + ESTR),  p2 = LD4(s, cb + 2*ESTR),  p3 = LD4(s, cb + 3*ESTR);
  float ym3[4] = {m3.x, m3.y, m3.z, m3.w}, ym2[4] = {m2.x, m2.y, m2.z, m2.w};
  float ym1[4] = {m1.x, m1.y, m1.z, m1.w}, yp1[4] = {p1.x, p1.y, p1.z, p1.w};
  float yp2[4] = {p2.x, p2.y, p2.z, p2.w}, yp3[4] = {p3.x, p3.y, p3.z, p3.w};
#pragma unroll
  for (int j = 0; j < 4; ++j) {
    float a1 = (w[3 + j] + w[5 + j]) + (ym1[j] + yp1[j]);
    float a2 = (w[2 + j] + w[6 + j]) + (ym2[j] + yp2[j]);
    float a3 = (w[1 + j] + w[7 + j]) + (ym3[j] + yp3[j]);
    ctr[j] = w[4 + j];
    lap[j] = fmaf(LC1, a1, fmaf(LC2, a2, fmaf(LC3, a3, LC0 * w[4 + j])));
  }
}

// One RK stage, x-quad blocked, fully b128-vectorized LDS traffic.
__device__ __forceinline__ void rk_stage(
    const float* __restrict__ su, const float* __restrict__ sv,
    const float* __restrict__ u0, const float* __restrict__ v0,
    float* __restrict__ du, float* __restrict__ dv,
    float* __restrict__ au, float* __restrict__ av,
    float mu_u, float mu_v, float na, float nb, float ff, float kf,
    float wk, float cs, bool writeStage, bool firstAcc, int tid)
{
  for (int i = tid; i < RROWS * QX; i += NTHR) {
    int ry  = i / QX;
    int q   = i - ry * QX;
    int ey  = ry + RB;
    int ex0 = q * 4;                       // 16B-aligned quad
    int cb  = ey * ESTR + ex0;

    float lu[4], lv[4], uc[4], vc[4], ku[4], kv[4];
    lap_quad(su, cb, lu, uc);
    lap_quad(sv, cb, lv, vc);
#pragma unroll
    for (int j = 0; j < 4; ++j) {
      float uv2 = uc[j] * vc[j] * vc[j];
      ku[j] = fmaf(mu_u, lu[j], fmaf(na, uv2, fmaf(-ff, uc[j], ff)));  // +f*(1-u)
      kv[j] = fmaf(mu_v, lv[j], fmaf(nb, uv2, -kf * vc[j]));           // -(k+f)*v
    }
    if (writeStage) {
      float4 bu = LD4(u0, cb), bv = LD4(v0, cb);
      float4 r;
      r.x = fmaf(cs, ku[0], bu.x); r.y = fmaf(cs, ku[1], bu.y);
      r.z = fmaf(cs, ku[2], bu.z); r.w = fmaf(cs, ku[3], bu.w);
      ST4(du, cb, r);
      r.x = fmaf(cs, kv[0], bv.x); r.y = fmaf(cs, kv[1], bv.y);
      r.z = fmaf(cs, kv[2], bv.z); r.w = fmaf(cs, kv[3], bv.w);
      ST4(dv, cb, r);
    }
    // central 64x64 is quad-aligned ([12,76)): uniform per-quad predicate
    int cy = ey - HALO, cx0 = ex0 - HALO;
    if ((unsigned)cy < (unsigned)TILE && (unsigned)cx0 < (unsigned)TILE) {
      int a = cy * TILE + cx0;
      float4 r;
      if (firstAcc) {
        r.x = wk * ku[0]; r.y = wk * ku[1]; r.z = wk * ku[2]; r.w = wk * ku[3];
        ST4(au, a, r);
        r.x = wk * kv[0]; r.y = wk * kv[1]; r.z = wk * kv[2]; r.w = wk * kv[3];
        ST4(av, a, r);
      } else {
        float4 o = LD4(au, a);
        r.x = fmaf(wk, ku[0], o.x); r.y = fmaf(wk, ku[1], o.y);
        r.z = fmaf(wk, ku[2], o.z); r.w = fmaf(wk, ku[3], o.w);
        ST4(au, a, r);
        o = LD4(av, a);
        r.x = fmaf(wk, kv[0], o.x); r.y = fmaf(wk, kv[1], o.y);
        r.z = fmaf(wk, kv[2], o.z); r.w = fmaf(wk, kv[3], o.w);
        ST4(av, a, r);
      }
    }
  }
}

__global__ __launch_bounds__(NTHR) void gray_scott_rk4(
    const float* __restrict__ h, float* __restrict__ out,
    const float* pCA, const float* pCB, const float* pNA,
    const float* pNB, const float* pF, const float* pK)
{
  extern __shared__ float4 smem4[];    // float4 -> guaranteed 16B base align
  float* smem = (float*)smem4;
  float* u0  = smem;                   // 88x88 each, bases all 16B-aligned
  float* v0  = u0  + EA;
  float* sAu = v0  + EA;
  float* sAv = sAu + EA;
  float* sBu = sAv + EA;
  float* sBv = sBu + EA;
  float* au  = sBv + EA;               // 64x64 RK accumulators
  float* av  = au  + TILE * TILE;

  const int tid = threadIdx.x;
  const int tx0 = blockIdx.x * TILE;
  const int ty0 = blockIdx.y * TILE;
  const int b   = blockIdx.z;

  const float* gu = h + ((size_t)b * 2 + 0) * (size_t)(1024 * 1024);
  const float* gv = h + ((size_t)b * 2 + 1) * (size_t)(1024 * 1024);

  // ---- async stage-in with wrap halo; pure-integer LDS addressing ----
  const uint32_t ldsb = (uint32_t)(uintptr_t)smem;       // hoisted once
  const uint32_t voff = ldsb + (uint32_t)(EA * sizeof(float));
  for (int i = tid; i < ESZ * ESZ; i += NTHR) {
    int ey = i / ESZ, ex = i - ey * ESZ;
    int gy = (ty0 + ey - HALO) & 1023;
    int gx = (tx0 + ex - HALO) & 1023;
    int go = (gy << 10) + gx;
    uint32_t lo = (uint32_t)((ey * ESTR + ex) * (int)sizeof(float));
    async_ld_b32(ldsb + lo, gu + go);
    async_ld_b32(voff + lo, gv + go);
  }

  // overlap scalar setup with async DMA
  float ca = *pCA, cb = *pCB, na = *pNA, nb = *pNB, ff = *pF, kk = *pK;
  float mu_u = 4e-5f / (1.0f + expf(-ca));
  float mu_v = 4e-5f / (1.0f + expf(-cb));
  float kf   = kk + ff;

  wait_async0();
  __syncthreads();

  // RK4 stages, ping-pong sB <-> sA
  rk_stage(u0,  v0,  u0, v0, sBu, sBv, au, av, mu_u, mu_v, na, nb, ff, kf,
           1.0f, 0.5f * kDT, true,  true,  tid);
  __syncthreads();
  rk_stage(sBu, sBv, u0, v0, sAu, sAv, au, av, mu_u, mu_v, na, nb, ff, kf,
           2.0f, 0.5f * kDT, true,  false, tid);
  __syncthreads();
  rk_stage(sAu, sAv, u0, v0, sBu, sBv, au, av, mu_u, mu_v, na, nb, ff, kf,
           2.0f, kDT,        true,  false, tid);
  __syncthreads();
  rk_stage(sBu, sBv, u0, v0, nullptr, nullptr, au, av, mu_u, mu_v, na, nb, ff, kf,
           1.0f, 0.0f,       false, false, tid);
  __syncthreads();

  // ---- combine + vectorized coalesced store ----
  float* ou = out + ((size_t)b * 2 + 0) * (size_t)(1024 * 1024);
  float* ov = out + ((size_t)b * 2 + 1) * (size_t)(1024 * 1024);
  const float w = kDT / 6.0f;
  for (int i = tid; i < (TILE * TILE) / 4; i += NTHR) {   // 2 iterations
    int oy  = i >> 4;
    int ox0 = (i & 15) << 2;
    int c   = (oy + HALO) * ESTR + (ox0 + HALO);
    int a   = oy * TILE + ox0;
    int g   = ((ty0 + oy) << 10) + (tx0 + ox0);
    float4 a4 = LD4(au, a), b4 = LD4(u0, c), r;
    r.x = fmaf(w, a4.x, b4.x); r.y = fmaf(w, a4.y, b4.y);
    r.z = fmaf(w, a4.z, b4.z); r.w = fmaf(w, a4.w, b4.w);
    *(float4*)(ou + g) = r;
    a4 = LD4(av, a); b4 = LD4(v0, c);
    r.x = fmaf(w, a4.x, b4.x); r.y = fmaf(w, a4.y, b4.y);
    r.z = fmaf(w, a4.z, b4.z); r.w = fmaf(w, a4.w, b4.w);
    *(float4*)(ov + g) = r;
  }
}

extern "C" void kernel_launch(void* const* d_in, const int* in_sizes, int n_in,
                              void* d_out, int out_size, void* d_ws, size_t ws_size,
                              hipStream_t stream) {
  (void)in_sizes; (void)n_in; (void)out_size; (void)d_ws; (void)ws_size;
  const float* h  = (const float*)d_in[0];
  const float* CA = (const float*)d_in[1];
  const float* CB = (const float*)d_in[2];
  const float* NA = (const float*)d_in[3];
  const float* NB = (const float*)d_in[4];
  const float* f  = (const float*)d_in[5];
  const float* k  = (const float*)d_in[6];
  float* out = (float*)d_out;

  dim3 grid(1024 / TILE, 1024 / TILE, 16);   // (16,16,16)
  dim3 block(NTHR);
  size_t shmem = (size_t)(6 * EA + 2 * TILE * TILE) * sizeof(float); // ~214 KB
  gray_scott_rk4<<<grid, block, shmem, stream>>>(h, out, CA, CB, NA, NB, f, k);
}